// TransformerLayer_8873402434050
// MI455X (gfx1250) — compile-verified
//
#include <hip/hip_runtime.h>
#include <hip/hip_bf16.h>

typedef float v2f __attribute__((ext_vector_type(2)));
typedef float v8f __attribute__((ext_vector_type(8)));

#define EPS 1e-5f
#define NEG_SLOPE 0.2f

// ---------- helpers ----------
__device__ __forceinline__ unsigned ord_encode(float f) {
    unsigned b = __float_as_uint(f);
    return (b & 0x80000000u) ? ~b : (b | 0x80000000u);
}
__device__ __forceinline__ float ord_decode(unsigned u) {
    unsigned b = (u & 0x80000000u) ? (u & 0x7FFFFFFFu) : ~u;
    return __uint_as_float(b);
}

// ---------- 1) self-loop attr accumulation: cnt[d] += 1, loopsum[d] += edge_attr[e] ----------
__global__ void k_deg_loop(const int* __restrict__ eidx, const float* __restrict__ eattr,
                           float* __restrict__ cnt, float* __restrict__ loopsum, int E) {
    int tid = blockIdx.x * blockDim.x + threadIdx.x;
    if (tid >= E * 8) return;
    int e = tid >> 3;
    int q = tid & 7;                       // 8 chunks of 4 floats = 32 attrs
    int d = eidx[E + e];                   // dst
    if (q == 0) atomicAdd(&cnt[d], 1.0f);
    const float4 v = *(const float4*)(eattr + (size_t)e * 32 + q * 4);
    float* dst = loopsum + (size_t)d * 32 + q * 4;
    atomicAdd(dst + 0, v.x);
    atomicAdd(dst + 1, v.y);
    atomicAdd(dst + 2, v.z);
    atomicAdd(dst + 3, v.w);
}

// ---------- 2) block GEMM: 16-row A panel staged in LDS once, 8 waves cover col tiles ----------
// C[m0..m0+15, :] = act(A_panel @ B + bias). K <= 512. One v_wmma_f32_16x16x4_f32 per K-step.
__global__ void k_gemm_blk(const float* __restrict__ A, const float* __restrict__ B,
                           const float* __restrict__ bias, float* __restrict__ C,
                           int K, int Nc, int relu) {
    __shared__ float sA[16 * 513];                 // 16 rows, stride K+1 (pad kills bank conflicts)
    const int tid  = threadIdx.x;
    const int wave = tid >> 5;
    const int lane = tid & 31;
    const int half = lane >> 4;                    // A/B VGPR pair: K = k0+half*2, +1
    const int c    = lane & 15;
    const long m0  = (long)blockIdx.x * 16;
    const int  st  = K + 1;

    // stage contiguous 16xK panel into LDS (rows are contiguous in global)
    for (int i = tid; i < 16 * K; i += 256) {
        int r = i / K, k = i - r * K;
        sA[r * st + k] = A[m0 * K + i];
    }
    __syncthreads();

    const int ntiles = Nc >> 4;
    for (int t = wave; t < ntiles; t += 8) {
        const int col = t * 16 + c;
        __builtin_prefetch(&B[col], 0, 1);         // warm the streamed B column
        v8f acc = {};
        for (int k0 = 0; k0 < K; k0 += 4) {
            const int ka = k0 + half * 2;
            v2f a, b;
            a.x = sA[c * st + ka];                 // row = c (A: lanes hold M = lane&15)
            a.y = sA[c * st + ka + 1];
            b.x = B[(long)ka * Nc + col];
            b.y = B[(long)(ka + 1) * Nc + col];
            acc = __builtin_amdgcn_wmma_f32_16x16x4_f32(false, a, false, b, (short)0, acc,
                                                        false, false);
        }
        const float bv = bias[col];
#pragma unroll
        for (int r = 0; r < 8; ++r) {
            long row = m0 + r + half * 8;          // C/D: VGPR r -> M=r (+8 upper half)
            float v = acc[r] + bv;
            if (relu) v = fmaxf(v, 0.0f);
            C[row * Nc + col] = v;
        }
    }
}

// ---------- 3) fused edge kernel: ep tile via WMMA, + gathers, leaky, att-dot, smax ----------
__global__ void k_edge_score(const int* __restrict__ eidx, const float* __restrict__ eattr,
                             const float* __restrict__ cnt, const float* __restrict__ loopsum,
                             const float* __restrict__ xl, const float* __restrict__ xr,
                             const float* __restrict__ We, const float* __restrict__ att,
                             float* __restrict__ score, unsigned* __restrict__ smaxU,
                             int E, int N) {
    __shared__ int sS[16], sD[16];
    const int lane = threadIdx.x;
    const int half = lane >> 4;
    const int c    = lane & 15;
    const long e0  = (long)blockIdx.x * 16;

    if (lane < 16) {
        long e = e0 + lane;
        int s, d;
        if (e < E) { s = eidx[e]; d = eidx[E + e]; }
        else       { s = d = (int)(e - E); }           // self loop
        sS[lane] = s;
        sD[lane] = d;
    }
    __syncthreads();

    // A fragments (this lane's edge row = e0 + c), all 8 K-steps of DE=32
    const long eA = e0 + c;
    float aK[8][2];
    if (eA < E) {
        const float* row = eattr + eA * 32;
#pragma unroll
        for (int s = 0; s < 8; ++s) {
            aK[s][0] = row[s * 4 + half * 2];
            aK[s][1] = row[s * 4 + half * 2 + 1];
        }
    } else {
        const long n = eA - E;
        const float inv = 1.0f / fmaxf(cnt[n], 1.0f);
        const float* row = loopsum + n * 32;
#pragma unroll
        for (int s = 0; s < 8; ++s) {
            aK[s][0] = row[s * 4 + half * 2] * inv;
            aK[s][1] = row[s * 4 + half * 2 + 1] * inv;
        }
    }

    // loop over heads: tile t covers cols [t*16, t*16+16) == head t (C==16)
    for (int t = 0; t < 8; ++t) {
        v8f acc = {};
#pragma unroll
        for (int s = 0; s < 8; ++s) {
            const int ka = s * 4 + half * 2;
            v2f a, b;
            a.x = aK[s][0];
            a.y = aK[s][1];
            b.x = We[(long)ka * 128 + t * 16 + c];
            b.y = We[(long)(ka + 1) * 128 + t * 16 + c];
            acc = __builtin_amdgcn_wmma_f32_16x16x4_f32(false, a, false, b, (short)0, acc,
                                                        false, false);
        }
        const int col  = t * 16 + c;
        const float av = att[t * 16 + c];
#pragma unroll
        for (int r = 0; r < 8; ++r) {
            const int el = r + half * 8;               // edge-in-tile for this acc element
            const int s  = sS[el];
            const int d  = sD[el];
            float v = acc[r] + xl[(long)s * 128 + col] + xr[(long)d * 128 + col];
            float m = (v > 0.0f) ? v : v * NEG_SLOPE;  // leaky relu
            float p = m * av;
            // reduce across the 16 lanes of this half (cols of head t)
            p += __shfl_xor(p, 1, 32);
            p += __shfl_xor(p, 2, 32);
            p += __shfl_xor(p, 4, 32);
            p += __shfl_xor(p, 8, 32);
            if (c == 0) {
                score[(e0 + el) * 8 + t] = p;
                atomicMax(&smaxU[(long)d * 8 + t], ord_encode(p));
            }
        }
    }
}

// ---------- 4) a = exp(score - smax[dst]); denom[dst] += a ----------
__global__ void k_exp_denom(const int* __restrict__ eidx, float* __restrict__ score,
                            const unsigned* __restrict__ smaxU, float* __restrict__ denom,
                            int E, int total) {
    int idx = blockIdx.x * blockDim.x + threadIdx.x;
    if (idx >= total * 8) return;
    int e = idx >> 3;
    int h = idx & 7;
    int d = (e < E) ? eidx[E + e] : (e - E);
    float a = __expf(score[idx] - ord_decode(smaxU[(long)d * 8 + h]));
    score[idx] = a;                                    // overwrite score with a
    atomicAdd(&denom[(long)d * 8 + h], a);
}

// ---------- 5) accum[dst] += xl[src] * alpha ----------
__global__ void k_aggregate(const int* __restrict__ eidx, const float* __restrict__ a,
                            const float* __restrict__ denom, const float* __restrict__ xl,
                            float* __restrict__ accum, int E, int total) {
    int idx = blockIdx.x * blockDim.x + threadIdx.x;
    if (idx >= total * 32) return;
    int e = idx >> 5;
    int q = idx & 31;                                   // 32 chunks of 4 -> 128 cols
    int h = q >> 2;
    int s, d;
    if (e < E) { s = eidx[e]; d = eidx[E + e]; }
    else       { s = d = e - E; }
    float alpha = a[(long)e * 8 + h] / denom[(long)d * 8 + h];
    const float4 xv = *(const float4*)(xl + (long)s * 128 + q * 4);
    float* dst = accum + (long)d * 128 + q * 4;
    atomicAdd(dst + 0, xv.x * alpha);
    atomicAdd(dst + 1, xv.y * alpha);
    atomicAdd(dst + 2, xv.z * alpha);
    atomicAdd(dst + 3, xv.w * alpha);
}

// ---------- 6) residual + LayerNorm (one wave per node, 4 elems/lane) ----------
__global__ void k_resid_ln(const float* __restrict__ resid, const float* __restrict__ add,
                           const float* __restrict__ extra_bias,   // may be null
                           const float* __restrict__ g, const float* __restrict__ beta,
                           float* __restrict__ out) {
    const long n = blockIdx.x;
    const int lane = threadIdx.x;
    const int j = lane * 4;
    float v[4];
    float s = 0.0f, s2 = 0.0f;
#pragma unroll
    for (int i = 0; i < 4; ++i) {
        float t = resid[n * 128 + j + i] + add[n * 128 + j + i];
        if (extra_bias) t += extra_bias[j + i];
        v[i] = t;
        s += t;
        s2 += t * t;
    }
#pragma unroll
    for (int m = 16; m >= 1; m >>= 1) {
        s  += __shfl_xor(s,  m, 32);
        s2 += __shfl_xor(s2, m, 32);
    }
    const float mean = s * (1.0f / 128.0f);
    const float var  = s2 * (1.0f / 128.0f) - mean * mean;
    const float inv  = rsqrtf(var + EPS);
#pragma unroll
    for (int i = 0; i < 4; ++i)
        out[n * 128 + j + i] = (v[i] - mean) * inv * g[j + i] + beta[j + i];
}

extern "C" void kernel_launch(void* const* d_in, const int* in_sizes, int n_in,
                              void* d_out, int out_size, void* d_ws, size_t ws_size,
                              hipStream_t stream) {
    const float* x     = (const float*)d_in[0];
    const int*   eidx  = (const int*)d_in[1];
    const float* eattr = (const float*)d_in[2];
    const float* Wl    = (const float*)d_in[3];
    const float* bl    = (const float*)d_in[4];
    const float* Wr    = (const float*)d_in[5];
    const float* br    = (const float*)d_in[6];
    const float* We    = (const float*)d_in[7];
    const float* att   = (const float*)d_in[8];
    const float* bias  = (const float*)d_in[9];
    const float* W1    = (const float*)d_in[10];
    const float* b1    = (const float*)d_in[11];
    const float* W2    = (const float*)d_in[12];
    const float* b2    = (const float*)d_in[13];
    const float* g1    = (const float*)d_in[14];
    const float* beta1 = (const float*)d_in[15];
    const float* g2    = (const float*)d_in[16];
    const float* beta2 = (const float*)d_in[17];

    const int N = in_sizes[0] / 128;       // 50000
    const int E = in_sizes[2] / 32;        // 800000
    const int T = E + N;                   // 850000 (divisible by 16)

    // workspace layout (floats)
    float*    w       = (float*)d_ws;
    float*    cnt     = w;                               // N
    float*    loopsum = cnt + N;                         // N*32
    unsigned* smaxU   = (unsigned*)(loopsum + (size_t)N * 32); // N*8
    float*    denom   = (float*)(smaxU + (size_t)N * 8); // N*8
    float*    accum   = denom + (size_t)N * 8;           // N*128
    float*    xl      = accum + (size_t)N * 128;         // N*128
    float*    xr      = xl + (size_t)N * 128;            // N*128
    float*    score   = xr + (size_t)N * 128;            // T*8
    float*    x1      = score + (size_t)T * 8;           // N*128
    float*    h1      = x1 + (size_t)N * 128;            // N*512
    float*    ff      = h1 + (size_t)N * 512;            // N*128

    // zero cnt|loopsum|smax|denom|accum (contiguous: N*177 floats)
    hipMemsetAsync(w, 0, (size_t)N * 177 * sizeof(float), stream);

    // 1) self-loop attr sums
    {
        int threads = E * 8;
        k_deg_loop<<<(threads + 255) / 256, 256, 0, stream>>>(eidx, eattr, cnt, loopsum, E);
    }
    // 2) xl = x@Wl+bl, xr = x@Wr+br  (WMMA f32, LDS-staged A panel)
    {
        k_gemm_blk<<<N / 16, 256, 0, stream>>>(x, Wl, bl, xl, 128, 128, 0);
        k_gemm_blk<<<N / 16, 256, 0, stream>>>(x, Wr, br, xr, 128, 128, 0);
    }
    // 3) fused edge: ep GEMM + gather + leaky + att-dot + segment-max
    k_edge_score<<<T / 16, 32, 0, stream>>>(eidx, eattr, cnt, loopsum, xl, xr, We, att,
                                            score, smaxU, E, N);
    // 4) exp + denom
    {
        int threads = T * 8;
        k_exp_denom<<<(threads + 255) / 256, 256, 0, stream>>>(eidx, score, smaxU, denom, E, T);
    }
    // 5) weighted scatter
    {
        int threads = T * 32;
        k_aggregate<<<(threads + 255) / 256, 256, 0, stream>>>(eidx, score, denom, xl, accum, E, T);
    }
    // 6) x1 = LN(x + accum + bias)
    k_resid_ln<<<N, 32, 0, stream>>>(x, accum, bias, g1, beta1, x1);
    // 7) FFN: h1 = relu(x1@W1+b1); ff = h1@W2+b2
    {
        k_gemm_blk<<<N / 16, 256, 0, stream>>>(x1, W1, b1, h1, 128, 512, 1);
        k_gemm_blk<<<N / 16, 256, 0, stream>>>(h1, W2, b2, ff, 512, 128, 0);
    }
    // 8) out = LN(x1 + ff)
    k_resid_ln<<<N, 32, 0, stream>>>(x1, ff, nullptr, g2, beta2, (float*)d_out);

    (void)n_in; (void)out_size; (void)ws_size;
}